// NerfHead_54898271977509
// MI455X (gfx1250) — compile-verified
//
#include <hip/hip_runtime.h>
#include <hip/hip_bf16.h>

typedef __attribute__((ext_vector_type(2))) float v2f;
typedef __attribute__((ext_vector_type(8))) float v8f;

#define N_RAYS    8192
#define N_INNER_  390
#define N_SAMP    416
#define NCLS      17

// float32-exact constants derived from the reference's numpy math
#define RADIUS_F      39.0f
#define BG_LEN_F      0.025641025789082050f   // float32((40-39)/39)
#define XY_EXT_F      1.0256410257890821f     // 1 + BG_LEN
#define Z_EXT_F       0.08000000119209290f    // float32(6.4000001/80)
#define SC_Z_F        2.2000000476837158f     // scene center z
#define ACT_SHIFT_F   (-13.815509557963774f)  // log(1/(1-1e-6)-1)
#define DIST_THR_F    0.0048717948725f        // (2+2*BG)/200*0.5*0.95
#define FAST_THR_F    1e-7f

__device__ __constant__ double FREQ_TAB[NCLS] = {
  1163161.0, 2309034.0, 188743.0, 2997643.0, 20317180.0, 852476.0,
  243808.0, 2457947.0, 497017.0, 2731022.0, 7224789.0, 214411435.0,
  5565043.0, 63191967.0, 76098082.0, 128860031.0, 141625221.0
};

// ---------------------------------------------------------------------------
// Wave32 sum-reduction using the CDNA5 matrix unit.
// A = ones(16x4) (per ISA layout: both A VGPRs = 1.0 in all lanes).
// B holds one value per lane in its first VGPR, zeros in the second, so the
// grand total of B equals the sum over lanes. With A = ones, every entry of
// D's row 0 is a column-sum of B; summing row 0 (lanes 0..15, VGPR0 of D,
// 4 xor-shuffles inside the 16-lane group) yields the full 32-lane sum,
// independent of B's exact (k,n) <-> (lane,vgpr) mapping.
// Requires EXEC all ones: called unconditionally by fully-populated waves.
// ---------------------------------------------------------------------------
__device__ __forceinline__ float wave_sum32(float x) {
  v2f a; a.x = 1.0f; a.y = 1.0f;
  v2f b; b.x = x;    b.y = 0.0f;
  v8f c = {0.f, 0.f, 0.f, 0.f, 0.f, 0.f, 0.f, 0.f};
  c = __builtin_amdgcn_wmma_f32_16x16x4_f32(false, a, false, b,
                                            (short)0, c, false, false);
  float s = c[0];
  s += __shfl_xor(s, 1);
  s += __shfl_xor(s, 2);
  s += __shfl_xor(s, 4);
  s += __shfl_xor(s, 8);
  return s;
}

extern "C" __global__ void nerf_zero(float* acc) {
  if (threadIdx.x < 8) acc[threadIdx.x] = 0.0f;
}

extern "C" __global__ __launch_bounds__(256)
void nerf_ray_kernel(const float* __restrict__ density,
                     const float* __restrict__ semantic,
                     const float* __restrict__ rays,
                     const float* __restrict__ bda,
                     float* __restrict__ acc)
{
  __shared__ float s_t[N_SAMP];
  __shared__ float s_bda[9];
  for (int i = threadIdx.x; i < N_SAMP; i += blockDim.x) {
    float t;
    if (i < N_INNER_) {
      t = (i + 0.5f) * (2.0f / (float)N_INNER_);
    } else {
      int j = i - N_INNER_;
      const float step = 0.984375f / 26.0f;          // (63/64)/26
      float l0 = 1.0f - (float)j       * step;
      float l1 = 1.0f - (float)(j + 1) * step;
      t = 1.0f / l0 + 1.0f / l1;                     // (b[j]+b[j+1])/2, b=2/l
    }
    s_t[i] = t;
  }
  if (threadIdx.x < 9) s_bda[threadIdx.x] = bda[threadIdx.x];
  __syncthreads();

  const int ray = blockIdx.x * blockDim.x + threadIdx.x;
  const float* __restrict__ r = rays + ray * 10;

  const float depth = r[2];
  const float gd    = (depth > 52.0f) ? 0.0f : depth;
  const bool  valid = gd > 0.0f;
  const float vf    = valid ? 1.0f : 0.0f;
  int gts = (int)r[3];
  gts = (gts < 0) ? 0 : (gts > NCLS - 1 ? NCLS - 1 : gts);

  // normalized origin / direction
  const float rox = (r[4] - 0.0f) * (1.0f / RADIUS_F);
  const float roy = (r[5] - 0.0f) * (1.0f / RADIUS_F);
  const float roz = (r[6] - SC_Z_F) * (1.0f / RADIUS_F);
  float dxn = r[7], dyn = r[8], dzn = r[9];
  {
    float inv = 1.0f / sqrtf(dxn * dxn + dyn * dyn + dzn * dzn);
    dxn *= inv; dyn *= inv; dzn *= inv;
  }

  const float SXY = 199.0f / (2.0f * XY_EXT_F);
  const float SZ  = 15.0f  / (2.0f * Z_EXT_F);

  float semacc[NCLS];
  #pragma unroll
  for (int c = 0; c < NCLS; ++c) semacc[c] = 0.0f;

  float T = 1.0f, cum = 0.0f;
  float cwsum = 0.0f, cwmsum = 0.0f;
  float sum_bi = 0.0f, sum_w2 = 0.0f, cnt_pk = 0.0f;
  float px = 0.0f, py = 0.0f, pz = 0.0f;

  #pragma unroll 1
  for (int s = 0; s < N_SAMP; ++s) {
    const float t = s_t[s];
    float qx = rox + dxn * t, qy = roy + dyn * t, qz = roz + dzn * t;
    const float nrm = sqrtf(qx * qx + qy * qy + qz * qz);
    const bool inner = nrm <= 1.0f;
    if (!inner) {
      const float inv = 1.0f / nrm;
      const float sc  = (1.0f + BG_LEN_F - BG_LEN_F * inv) * inv;
      qx *= sc; qy *= sc; qz *= sc;
    }
    const float bx = s_bda[0]*qx + s_bda[1]*qy + s_bda[2]*qz;
    const float by = s_bda[3]*qx + s_bda[4]*qy + s_bda[5]*qz;
    const float bz = s_bda[6]*qx + s_bda[7]*qy + s_bda[8]*qz;

    bool keep;
    if (s == 0) {
      keep = inner;
    } else {
      const float ddx = bx - px, ddy = by - py, ddz = bz - pz;
      cum += sqrtf(ddx * ddx + ddy * ddy + ddz * ddz);
      const bool over = cum > DIST_THR_F;
      if (over) cum = 0.0f;
      keep = inner || over;
    }
    px = bx; py = by; pz = bz;

    float a = 0.0f;
    float cw8[8] = {0.f,0.f,0.f,0.f,0.f,0.f,0.f,0.f};
    int   off8[8] = {0,0,0,0,0,0,0,0};
    if (keep && valid) {
      const float gx = (bx + XY_EXT_F) * SXY;
      const float gy = (by + XY_EXT_F) * SXY;
      const float gz = (bz + Z_EXT_F)  * SZ;
      const float fx = floorf(gx), fy = floorf(gy), fz = floorf(gz);
      const int ix = (int)fx, iy = (int)fy, iz = (int)fz;
      const float tx = gx - fx, ty = gy - fy, tz = gz - fz;
      const float wxa[2] = {1.0f - tx, tx};
      const float wya[2] = {1.0f - ty, ty};
      const float wza[2] = {1.0f - tz, tz};
      const int xia[2] = {ix, ix + 1};
      const int yia[2] = {iy, iy + 1};
      const int zia[2] = {iz, iz + 1};
      float dens = 0.0f;
      int k = 0;
      #pragma unroll
      for (int dX = 0; dX < 2; ++dX)
        #pragma unroll
        for (int dY = 0; dY < 2; ++dY)
          #pragma unroll
          for (int dZ = 0; dZ < 2; ++dZ) {
            const int X = xia[dX], Y = yia[dY], Z = zia[dZ];
            const bool v = (X >= 0) & (X < 200) & (Y >= 0) & (Y < 200) &
                           (Z >= 0) & (Z < 16);
            const int Xc = min(max(X, 0), 199);
            const int Yc = min(max(Y, 0), 199);
            const int Zc = min(max(Z, 0), 15);
            const float w = wxa[dX] * wya[dY] * wza[dZ] * (v ? 1.0f : 0.0f);
            const int off = (Xc * 200 + Yc) * 16 + Zc;
            cw8[k] = w; off8[k] = off; ++k;
            dens += w * density[off];
          }
      const float e  = expf(dens + ACT_SHIFT_F);
      const float al = 1.0f - 1.0f / sqrtf(1.0f + e);
      a = (al > FAST_THR_F) ? al : 0.0f;
    }

    const float wgt = a * T;
    T *= (1.0f - a);
    const bool pk = wgt > FAST_THR_F;
    const float w = pk ? wgt : 0.0f;
    cnt_pk += pk ? 1.0f : 0.0f;

    const float m = 1.0f - 1.0f / (1.0f + t);
    sum_bi += 2.0f * w * (m * cwsum - cwmsum);   // exclusive prefix sums
    sum_w2 += w * w;
    cwsum  += w;
    cwmsum += w * m;

    if (w > 0.0f) {
      #pragma unroll
      for (int k2 = 0; k2 < 8; ++k2) {
        const float wk = w * cw8[k2];
        if (wk != 0.0f) {
          const float* __restrict__ sp = semantic + off8[k2] * NCLS;
          #pragma unroll
          for (int c = 0; c < NCLS; ++c) semacc[c] += wk * sp[c];
        }
      }
    }
  }

  // entropy of background probability
  float p = fminf(fmaxf(T, 1e-6f), 1.0f - 1e-6f);
  const float ent = -(p * logf(p) + (1.0f - p) * logf(1.0f - p)) * vf;

  // weighted NLL (constant-index loops only; keep semacc in registers)
  float mx = semacc[0];
  #pragma unroll
  for (int c = 1; c < NCLS; ++c) mx = fmaxf(mx, semacc[c]);
  float se = 0.0f;
  #pragma unroll
  for (int c = 0; c < NCLS; ++c) se += expf(semacc[c] - mx);
  const float lse = mx + logf(se);
  float sg = 0.0f;
  #pragma unroll
  for (int c = 0; c < NCLS; ++c) sg = (c == gts) ? semacc[c] : sg;
  const float nll = lse - sg;
  const float wy  = vf * (float)(1.0 / log(FREQ_TAB[gts] + 0.001));

  // --- WMMA-based wave32 reductions of the seven loss partials ---
  const float r0 = wave_sum32(wy * nll);
  const float r1 = wave_sum32(wy);
  const float r2 = wave_sum32(ent);
  const float r3 = wave_sum32(vf);
  const float r4 = wave_sum32(sum_bi);
  const float r5 = wave_sum32(sum_w2);
  const float r6 = wave_sum32(cnt_pk);

  if ((threadIdx.x & 31) == 0) {
    atomicAdd(&acc[0], r0);
    atomicAdd(&acc[1], r1);
    atomicAdd(&acc[2], r2);
    atomicAdd(&acc[3], r3);
    atomicAdd(&acc[4], r4);
    atomicAdd(&acc[5], r5);
    atomicAdd(&acc[6], r6);
  }
}

extern "C" __global__ void nerf_finalize(const float* __restrict__ acc,
                                         float* __restrict__ out)
{
  if (threadIdx.x == 0 && blockIdx.x == 0) {
    const float nv    = fmaxf(acc[3], 1.0f);
    const float lsem  = 1.0f  * acc[0] / fmaxf(acc[1], 1e-12f);
    const float lent  = 0.01f * acc[2] / nv;
    const float nmax  = fmaxf(acc[6], 1.0f);
    const float ldist = 0.01f * (acc[4] + (1.0f / 3.0f) * (1.0f / nmax) * acc[5]) / nv;
    out[0] = lsem;
    out[1] = lent;
    out[2] = ldist;
  }
}

extern "C" void kernel_launch(void* const* d_in, const int* in_sizes, int n_in,
                              void* d_out, int out_size, void* d_ws, size_t ws_size,
                              hipStream_t stream) {
  const float* density  = (const float*)d_in[0];   // (1,200,200,16)
  const float* semantic = (const float*)d_in[1];   // (1,200,200,16,17)
  const float* rays     = (const float*)d_in[2];   // (1,8192,10)
  const float* bda      = (const float*)d_in[3];   // (1,3,3)
  float* acc = (float*)d_ws;

  nerf_zero<<<1, 32, 0, stream>>>(acc);
  nerf_ray_kernel<<<N_RAYS / 256, 256, 0, stream>>>(density, semantic, rays,
                                                    bda, acc);
  nerf_finalize<<<1, 1, 0, stream>>>(acc, (float*)d_out);
}